// DetectionPostprocess_6700148982188
// MI455X (gfx1250) — compile-verified
//
#include <hip/hip_runtime.h>
#include <math.h>
#include <stdint.h>

#define TPB 256
#define NSEL 20
#define NCAND (TPB * NSEL)
// level sizes: 32^3=32768, 16^3=4096, 8^3=512 ; total 37376 ; 180*8 out per image
#define N0 32768
#define N1 4096
#define N2 512
#define SH1 32768
#define SH2 36864
#define OUTROWS 180

__device__ __forceinline__ float sigm(float x) { return 1.0f / (1.0f + expf(-x)); }

// Monotonic 64-bit key: (score desc, idx asc) <=> unsigned key desc.
// Valid because sigmoid scores here are strictly positive normals, whose IEEE
// bit patterns are order-isomorphic to their float values.
__device__ __forceinline__ uint64_t make_key(float s, uint32_t n) {
    return ((uint64_t)__float_as_uint(s) << 32) | (uint64_t)(0xFFFFFFFFu - n);
}

// Register-resident top-20 list of packed keys (kept fully unrolled -> VGPRs).
struct TopK {
    uint64_t e[NSEL];
    uint64_t wk; int wp;   // current worst key and its slot
    __device__ void init() {
#pragma unroll
        for (int k = 0; k < NSEL; ++k) e[k] = 0ull;
        wk = 0ull; wp = 0;
    }
    __device__ void consider(uint64_t key) {
        if (key > wk) {
#pragma unroll
            for (int k = 0; k < NSEL; ++k) if (k == wp) e[k] = key;
            uint64_t m = e[0]; int p = 0;
#pragma unroll
            for (int k = 1; k < NSEL; ++k) if (e[k] < m) { m = e[k]; p = k; }
            wk = m; wp = p;
        }
    }
};

__global__ __launch_bounds__(TPB) void det_post_kernel(
    const float* __restrict__ cls0, const float* __restrict__ shp0, const float* __restrict__ off0,
    const float* __restrict__ cls1, const float* __restrict__ shp1, const float* __restrict__ off1,
    const float* __restrict__ cls2, const float* __restrict__ shp2, const float* __restrict__ off2,
    float* __restrict__ out)
{
    const int b   = blockIdx.x;
    const int tid = threadIdx.x;

    __shared__ uint64_t lds_key[NCAND];          // 40 KB
    __shared__ uint64_t sel_key[NSEL];
    __shared__ float    sel_sc[NSEL];
    __shared__ float    boxes[NSEL][6];
    __shared__ float    iou[NSEL * NSEL];
    __shared__ float    staging[NSEL][8];
    __shared__ int      kcount_s;

    // ---- Phase 1: per-thread top-20 over all 37376 anchors (B128 loads) ----
    TopK tl; tl.init();
    {
        const float4* c0 = reinterpret_cast<const float4*>(cls0 + (size_t)b * N0);
        for (int i = tid; i < N0 / 4; i += TPB) {
            __builtin_prefetch(c0 + i + 4 * TPB, 0, 0);      // global_prefetch_b8
            const float4 v = c0[i];
            const uint32_t n = (uint32_t)i * 4u;
            tl.consider(make_key(sigm(v.x), n + 0u));
            tl.consider(make_key(sigm(v.y), n + 1u));
            tl.consider(make_key(sigm(v.z), n + 2u));
            tl.consider(make_key(sigm(v.w), n + 3u));
        }
        const float4* c1 = reinterpret_cast<const float4*>(cls1 + (size_t)b * N1);
        for (int i = tid; i < N1 / 4; i += TPB) {
            const float4 v = c1[i];
            const uint32_t n = SH1 + (uint32_t)i * 4u;
            tl.consider(make_key(sigm(v.x), n + 0u));
            tl.consider(make_key(sigm(v.y), n + 1u));
            tl.consider(make_key(sigm(v.z), n + 2u));
            tl.consider(make_key(sigm(v.w), n + 3u));
        }
        const float4* c2 = reinterpret_cast<const float4*>(cls2 + (size_t)b * N2);
        for (int i = tid; i < N2 / 4; i += TPB) {             // only 128 threads active
            const float4 v = c2[i];
            const uint32_t n = SH2 + (uint32_t)i * 4u;
            tl.consider(make_key(sigm(v.x), n + 0u));
            tl.consider(make_key(sigm(v.y), n + 1u));
            tl.consider(make_key(sigm(v.z), n + 2u));
            tl.consider(make_key(sigm(v.w), n + 3u));
        }
    }
#pragma unroll
    for (int k = 0; k < NSEL; ++k) lds_key[tid * NSEL + k] = tl.e[k];
    __syncthreads();

    // ---- Phase 2: exact global top-20 merge (single wave32, u64 max) -------
    for (int r = 0; r < NSEL; ++r) {
        if (tid < 32) {
            uint64_t vK = 0ull; int vP = 0;
            for (int ei = tid; ei < NCAND; ei += 32) {
                const uint64_t k = lds_key[ei];
                if (k > vK) { vK = k; vP = ei; }
            }
            for (int off = 16; off > 0; off >>= 1) {          // wave32 xor-reduce
                const uint64_t oK = (uint64_t)__shfl_xor((unsigned long long)vK, off, 32);
                const int      oP = __shfl_xor(vP, off, 32);
                if (oK > vK) { vK = oK; vP = oP; }
            }
            if (tid == 0) {
                sel_key[r] = vK;
                lds_key[vP] = 0ull;                           // consume winner
            }
        }
        __syncthreads();
    }

    // ---- Phase 3: decode the 20 candidate boxes ----------------------------
    if (tid < NSEL) {
        const uint64_t key = sel_key[tid];
        const float s = __uint_as_float((uint32_t)(key >> 32));
        const int   n = (int)(0xFFFFFFFFu - (uint32_t)(key & 0xFFFFFFFFull));
        sel_sc[tid] = s;

        const float *shp, *off;
        int D, ln; float stride;
        if (n < SH1)      { D = 32; stride =  4.0f; ln = n;       shp = shp0 + (size_t)b * 3 * N0; off = off0 + (size_t)b * 3 * N0; }
        else if (n < SH2) { D = 16; stride =  8.0f; ln = n - SH1; shp = shp1 + (size_t)b * 3 * N1; off = off1 + (size_t)b * 3 * N1; }
        else              { D =  8; stride = 16.0f; ln = n - SH2; shp = shp2 + (size_t)b * 3 * N2; off = off2 + (size_t)b * 3 * N2; }
        const int D2 = D * D, D3 = D2 * D;
        const int z = ln / D2, rem = ln % D2, y = rem / D, x = rem % D;
        const float oz = off[0 * D3 + ln], oy = off[1 * D3 + ln], ox = off[2 * D3 + ln];
        const float sz = shp[0 * D3 + ln], sy = shp[1 * D3 + ln], sx = shp[2 * D3 + ln];
        boxes[tid][0] = ((float)z + oz) * stride;
        boxes[tid][1] = ((float)y + oy) * stride;
        boxes[tid][2] = ((float)x + ox) * stride;
        boxes[tid][3] = sz; boxes[tid][4] = sy; boxes[tid][5] = sx;
    }
    __syncthreads();

    // ---- Phase 4a: pairwise 3D IoU (400 pairs in parallel) -----------------
    for (int p = tid; p < NSEL * NSEL; p += TPB) {
        const int i = p / NSEL, j = p % NSEL;
        const float iz = boxes[i][0], iy = boxes[i][1], ix = boxes[i][2];
        const float jz = boxes[j][0], jy = boxes[j][1], jx = boxes[j][2];
        const float id_ = fmaxf(boxes[i][3], 0.0f), ih = fmaxf(boxes[i][4], 0.0f), iw = fmaxf(boxes[i][5], 0.0f);
        const float jd  = fmaxf(boxes[j][3], 0.0f), jh = fmaxf(boxes[j][4], 0.0f), jw = fmaxf(boxes[j][5], 0.0f);
        const float t0 = fmaxf(fminf(iz + 0.5f * id_, jz + 0.5f * jd) - fmaxf(iz - 0.5f * id_, jz - 0.5f * jd), 0.0f);
        const float t1 = fmaxf(fminf(iy + 0.5f * ih , jy + 0.5f * jh) - fmaxf(iy - 0.5f * ih , jy - 0.5f * jh), 0.0f);
        const float t2 = fmaxf(fminf(ix + 0.5f * iw , jx + 0.5f * jw) - fmaxf(ix - 0.5f * iw , jx - 0.5f * jw), 0.0f);
        const float inter = t0 * t1 * t2;
        const float vi = id_ * ih * iw, vj = jd * jh * jw;
        iou[p] = inter / (vi + vj - inter + 1e-8f);
    }
    __syncthreads();

    // ---- Phase 4b: greedy NMS (inherently serial, 20 steps) ----------------
    if (tid == 0) {
        bool keep[NSEL];
        int kc = 0;
        for (int i = 0; i < NSEL; ++i) {
            const bool v = sel_sc[i] > 0.15f;
            bool sup = false;
            for (int j = 0; j < i; ++j)
                if (keep[j] && iou[i * NSEL + j] > 0.05f) sup = true;
            keep[i] = v && !sup;
            if (keep[i]) {
                staging[kc][0] = 1.0f;
                staging[kc][1] = sel_sc[i];
                for (int c = 0; c < 6; ++c) staging[kc][2 + c] = boxes[i][c];
                ++kc;
            }
        }
        kcount_s = kc;
    }
    __syncthreads();

    // ---- Phase 5: write dets [180,8] (race-free, deterministic) ------------
    float* ob = out + (size_t)b * OUTROWS * 8;
    const int kc = kcount_s;
    for (int e = tid; e < OUTROWS * 8; e += TPB) {
        const int row = e >> 3, col = e & 7;
        ob[e] = (row < kc) ? staging[row][col] : -1.0f;
    }
}

extern "C" void kernel_launch(void* const* d_in, const int* in_sizes, int n_in,
                              void* d_out, int out_size, void* d_ws, size_t ws_size,
                              hipStream_t stream) {
    const float* cls0 = (const float*)d_in[0];
    const float* shp0 = (const float*)d_in[1];
    const float* off0 = (const float*)d_in[2];
    const float* cls1 = (const float*)d_in[3];
    const float* shp1 = (const float*)d_in[4];
    const float* off1 = (const float*)d_in[5];
    const float* cls2 = (const float*)d_in[6];
    const float* shp2 = (const float*)d_in[7];
    const float* off2 = (const float*)d_in[8];
    float* out = (float*)d_out;

    det_post_kernel<<<dim3(64), dim3(TPB), 0, stream>>>(
        cls0, shp0, off0, cls1, shp1, off1, cls2, shp2, off2, out);
}